// TargetRanker_56556129354064
// MI455X (gfx1250) — compile-verified
//
#include <hip/hip_runtime.h>

// ---------------------------------------------------------------------------
// GCN (3x GCNConv + ReLU) + global mean pool, MI455X / gfx1250 (wave32, WMMA)
// GEMM uses V_WMMA_F32_16X16X4_F32 (full fp32 precision on the matrix pipe).
// Scatter phase uses hardware global_atomic_add_f32 (L2-resident buffers).
// ---------------------------------------------------------------------------

typedef float v2f __attribute__((ext_vector_type(2)));
typedef float v8f __attribute__((ext_vector_type(8)));

#define HID 128
#define NGRAPH 64

// LDS strides chosen for conflict-free access patterns (64 banks x 4B):
//  sA stride 132: addr = 132*m + k -> bank (4m+k)%64, distinct across m=0..15
//  sW stride 136: rows k and k+2 land 16 banks apart -> lanes 0-15 vs 16-31 disjoint
#define SA_STRIDE 132
#define SW_STRIDE 136

__global__ void fill_kernel(float* __restrict__ p, float v, int n) {
    int i = blockIdx.x * blockDim.x + threadIdx.x;
    if (i < n) p[i] = v;
}

__global__ void deg_edge_kernel(const long long* __restrict__ dst,
                                float* __restrict__ deg, int nedges) {
    int i = blockIdx.x * blockDim.x + threadIdx.x;
    if (i < nedges) unsafeAtomicAdd(&deg[(int)dst[i]], 1.0f);
}

__global__ void rsqrt_kernel(float* __restrict__ p, int n) {
    int i = blockIdx.x * blockDim.x + threadIdx.x;
    if (i < n) p[i] = rsqrtf(p[i]);
}

// One block = 256 threads = 8 waves. Block b computes rows [16b,16b+16) of
// hw = h_in @ W; wave w computes output columns [16w,16w+16). Epilogue fuses
// the GCN self-loop + bias init:  agg = hw * dinv[row]^2 + bias[col].
__global__ __launch_bounds__(256) void gemm_kernel(
    const float* __restrict__ hin, const float* __restrict__ W,
    const float* __restrict__ bias, const float* __restrict__ dinv,
    float* __restrict__ hw, float* __restrict__ agg) {
    __shared__ float sW[HID * SW_STRIDE];
    __shared__ float sA[16 * SA_STRIDE];

    const int tid  = threadIdx.x;
    const int lane = tid & 31;
    const int wave = tid >> 5;            // 0..7 -> N-tile
    const int row0 = blockIdx.x * 16;

    // Stage W (128x128 f32 = 64KB) into LDS, padded stride.
    for (int i = tid * 4; i < HID * HID; i += 256 * 4) {
        float4 v = *(const float4*)(W + i);
        int k = i >> 7, n = i & 127;
        *(float4*)&sW[k * SW_STRIDE + n] = v;
    }
    // Stage A tile (16x128 f32 = 8KB) into LDS, padded stride.
    for (int i = tid * 4; i < 16 * HID; i += 256 * 4) {
        float4 v = *(const float4*)(hin + (size_t)row0 * HID + i);
        int r = i >> 7, c = i & 127;
        *(float4*)&sA[r * SA_STRIDE + c] = v;
    }
    __syncthreads();

    const int m  = lane & 15;   // A row within tile / C column
    const int hi = lane >> 4;   // 0: K pair {0,1}, 1: K pair {2,3}
    const int n  = wave * 16 + m;  // B column

    v8f acc = {};
    #pragma unroll
    for (int k0 = 0; k0 < HID; k0 += 4) {
        const int ka = k0 + 2 * hi;
        v2f a, b;
        a.x = sA[m * SA_STRIDE + ka];
        a.y = sA[m * SA_STRIDE + ka + 1];
        b.x = sW[ka * SW_STRIDE + n];
        b.y = sW[(ka + 1) * SW_STRIDE + n];
        // D = A(16x4) x B(4x16) + C, fp32 throughout
        acc = __builtin_amdgcn_wmma_f32_16x16x4_f32(
            false, a, false, b, (short)0, acc, false, false);
    }

    const float bcol = bias[n];
    #pragma unroll
    for (int r = 0; r < 8; ++r) {
        const int grow = row0 + r + 8 * hi;     // C layout: VGPR r -> M = r + 8*hi
        const size_t idx = (size_t)grow * HID + n;
        const float v  = acc[r];
        const float di = dinv[grow];
        hw[idx]  = v;
        agg[idx] = v * di * di + bcol;          // self-loop term + bias
    }
}

// One wave per edge; each lane handles 4 consecutive features (float4 gather,
// 4 hardware f32 atomics into the destination row).
__global__ __launch_bounds__(256) void scatter_kernel(
    const long long* __restrict__ srcIdx, const long long* __restrict__ dstIdx,
    const float* __restrict__ hw, const float* __restrict__ dinv,
    float* __restrict__ agg, int nedges) {
    const int e = blockIdx.x * 8 + (threadIdx.x >> 5);
    if (e >= nedges) return;
    const int lane = threadIdx.x & 31;
    const int s = (int)srcIdx[e];
    const int d = (int)dstIdx[e];
    const float nrm = dinv[s] * dinv[d];
    const float4 v = *(const float4*)(hw + (size_t)s * HID + lane * 4);
    float* out = agg + (size_t)d * HID + lane * 4;
    unsafeAtomicAdd(out + 0, v.x * nrm);
    unsafeAtomicAdd(out + 1, v.y * nrm);
    unsafeAtomicAdd(out + 2, v.z * nrm);
    unsafeAtomicAdd(out + 3, v.w * nrm);
}

__global__ void relu_kernel(const float* __restrict__ in, float* __restrict__ out,
                            int n4) {
    int i = blockIdx.x * blockDim.x + threadIdx.x;
    if (i < n4) {
        float4 v = ((const float4*)in)[i];
        v.x = fmaxf(v.x, 0.0f); v.y = fmaxf(v.y, 0.0f);
        v.z = fmaxf(v.z, 0.0f); v.w = fmaxf(v.w, 0.0f);
        ((float4*)out)[i] = v;
    }
}

__global__ void pool_count_kernel(const long long* __restrict__ batch,
                                  float* __restrict__ counts, int n) {
    int i = blockIdx.x * blockDim.x + threadIdx.x;
    if (i < n) unsafeAtomicAdd(&counts[(int)batch[i]], 1.0f);
}

// One wave per node: lane handles 4 features -> atomic sum into out[graph].
__global__ void pool_sum_kernel(const long long* __restrict__ batch,
                                const float* __restrict__ h,
                                float* __restrict__ out, int n) {
    int t = blockIdx.x * blockDim.x + threadIdx.x;
    int node = t >> 5;
    if (node >= n) return;
    int c = (t & 31) * 4;
    int g = (int)batch[node];
    const float4 v = *(const float4*)(h + (size_t)node * HID + c);
    float* o = out + (size_t)g * HID + c;
    unsafeAtomicAdd(o + 0, v.x);
    unsafeAtomicAdd(o + 1, v.y);
    unsafeAtomicAdd(o + 2, v.z);
    unsafeAtomicAdd(o + 3, v.w);
}

__global__ void pool_div_kernel(float* __restrict__ out,
                                const float* __restrict__ counts, int n) {
    int i = blockIdx.x * blockDim.x + threadIdx.x;
    if (i < n) out[i] /= fmaxf(counts[i >> 7], 1.0f);
}

extern "C" void kernel_launch(void* const* d_in, const int* in_sizes, int n_in,
                              void* d_out, int out_size, void* d_ws, size_t ws_size,
                              hipStream_t stream) {
    const float*     x     = (const float*)d_in[0];
    const long long* ei    = (const long long*)d_in[1];   // [2, E] int64
    const long long* batch = (const long long*)d_in[2];   // [N] int64
    const float* W0 = (const float*)d_in[3];
    const float* b0 = (const float*)d_in[4];
    const float* W1 = (const float*)d_in[5];
    const float* b1 = (const float*)d_in[6];
    const float* W2 = (const float*)d_in[7];
    const float* b2 = (const float*)d_in[8];
    float* out = (float*)d_out;

    const int N = in_sizes[0] / HID;        // 50000
    const int E = in_sizes[1] / 2;          // 800000
    const long long* src = ei;
    const long long* dst = ei + E;

    // Workspace carve-up (floats): dinv[N] | bufA[N*128] | bufB[N*128] | agg[N*128] | counts[64]
    const size_t feat = (size_t)N * HID;
    float* dinv   = (float*)d_ws;
    float* bufA   = dinv + (((size_t)N + 255) & ~(size_t)255);
    float* bufB   = bufA + feat;
    float* aggb   = bufB + feat;
    float* counts = aggb + feat;

    const int B256 = 256;
    // 1) degrees (self-loop => start at 1), then dinv = rsqrt(deg)
    fill_kernel<<<(N + B256 - 1) / B256, B256, 0, stream>>>(dinv, 1.0f, N);
    deg_edge_kernel<<<(E + B256 - 1) / B256, B256, 0, stream>>>(dst, dinv, E);
    rsqrt_kernel<<<(N + B256 - 1) / B256, B256, 0, stream>>>(dinv, N);

    const int gemmGrid    = (N + 15) / 16;      // 3125
    const int scatterGrid = (E + 7) / 8;        // 100000
    const int n4          = (int)(feat / 4);    // 1.6M float4
    const int reluGrid    = (n4 + B256 - 1) / B256;

    const float* cur = x;
    const float* Ws[3] = {W0, W1, W2};
    const float* bs[3] = {b0, b1, b2};
    for (int l = 0; l < 3; ++l) {
        gemm_kernel<<<gemmGrid, B256, 0, stream>>>(cur, Ws[l], bs[l], dinv, bufA, aggb);
        scatter_kernel<<<scatterGrid, B256, 0, stream>>>(src, dst, bufA, dinv, aggb, E);
        relu_kernel<<<reluGrid, B256, 0, stream>>>(aggb, bufB, n4);
        cur = bufB;
    }

    // 2) global mean pool
    fill_kernel<<<(out_size + B256 - 1) / B256, B256, 0, stream>>>(out, 0.0f, out_size);
    fill_kernel<<<1, NGRAPH, 0, stream>>>(counts, 0.0f, NGRAPH);
    pool_count_kernel<<<(N + B256 - 1) / B256, B256, 0, stream>>>(batch, counts, N);
    pool_sum_kernel<<<(N * 32 + B256 - 1) / B256, B256, 0, stream>>>(batch, bufB, out, N);
    pool_div_kernel<<<(out_size + B256 - 1) / B256, B256, 0, stream>>>(out, counts, out_size);
}